// HDNet_52707838657074
// MI455X (gfx1250) — compile-verified
//
#include <hip/hip_runtime.h>
#include <stdint.h>

typedef __attribute__((ext_vector_type(2))) float v2f;
typedef __attribute__((ext_vector_type(8))) float v8f;

#define BM 128
#define BN 64
#define BK 16
#define PAD 20   // padded LDS row stride (floats): 80B, 16B-aligned, bank-conflict-free

// -------------------- CDNA5 async global->LDS copy helpers --------------------
__device__ __forceinline__ void async_b128(uint32_t lds_off, const float* g) {
    asm volatile("global_load_async_to_lds_b128 %0, %1, off"
                 :: "v"(lds_off), "v"(g) : "memory");
}
__device__ __forceinline__ void async_b32(uint32_t lds_off, const float* g) {
    asm volatile("global_load_async_to_lds_b32 %0, %1, off"
                 :: "v"(lds_off), "v"(g) : "memory");
}
__device__ __forceinline__ void wait_async0() {
    asm volatile("s_wait_asynccnt 0" ::: "memory");
}
__device__ __forceinline__ uint32_t lds_off_of(const void* p) {
    // LDS generic addresses carry the LDS byte offset in the low 32 bits
    return (uint32_t)(uintptr_t)p;
}

// ---------------------------------------------------------------------------
// FP32 WMMA GEMM: C[M,N] = A[M,K] @ B[K,N], row-major. M%128==0, N%64==0, K%16==0.
// Block = 256 threads (8 wave32) computing a 128x64 macro-tile.
// K-panels of 16 are DMA'd into double-buffered LDS with async-to-LDS loads;
// each wave runs 16x V_WMMA_F32_16X16X4_F32 per panel from ds_load fragments.
// ---------------------------------------------------------------------------
__global__ __launch_bounds__(256) void gemm_wmma_f32(
    const float* __restrict__ A, const float* __restrict__ B,
    float* __restrict__ C, int M, int N, int K)
{
    __shared__ float As[2][BM * PAD];   // row-major [row][k]
    __shared__ float Bs[2][BN * PAD];   // transposed [col][k]

    const int tid    = threadIdx.x;
    const int lane   = tid & 31;
    const int wave   = tid >> 5;
    const int laneHi = lane >> 4;       // 0 or 1
    const int lanLo  = lane & 15;
    const int mbase  = blockIdx.y * BM;
    const int nbase  = blockIdx.x * BN;

    v8f acc0 = {}, acc1 = {}, acc2 = {}, acc3 = {};

    // A staging: 128 rows x 16 floats; thread t -> row t/2, 8 floats (2x b128)
    const int  arow  = tid >> 1;
    const int  akoff = (tid & 1) * 8;

    auto stage = [&](int buf, int k0) {
        const float* ga = A + (size_t)(mbase + arow) * K + k0 + akoff;
        uint32_t la = lds_off_of(&As[buf][arow * PAD + akoff]);
        async_b128(la,      ga);
        async_b128(la + 16, ga + 4);
        // B staging (transposed into LDS): 16 k-rows x 64 cols, 4x b32 per thread
#pragma unroll
        for (int i = 0; i < 4; ++i) {
            int e = i * 256 + tid;
            int r = e >> 6;          // k within panel
            int c = e & 63;          // column within tile
            const float* gb = B + (size_t)(k0 + r) * N + nbase + c;
            async_b32(lds_off_of(&Bs[buf][c * PAD + r]), gb);
        }
    };

    auto compute = [&](int buf) {
        const float* Ab = &As[buf][(wave * 16 + lanLo) * PAD + 2 * laneHi];
        const float* Bb = &Bs[buf][lanLo * PAD + 2 * laneHi];
#pragma unroll
        for (int kk = 0; kk < BK; kk += 4) {
            v2f af = *(const v2f*)(Ab + kk);
            v2f b0 = *(const v2f*)(Bb + 0 * 16 * PAD + kk);
            v2f b1 = *(const v2f*)(Bb + 1 * 16 * PAD + kk);
            v2f b2 = *(const v2f*)(Bb + 2 * 16 * PAD + kk);
            v2f b3 = *(const v2f*)(Bb + 3 * 16 * PAD + kk);
            acc0 = __builtin_amdgcn_wmma_f32_16x16x4_f32(false, af, false, b0,
                                                         (short)0, acc0, false, false);
            acc1 = __builtin_amdgcn_wmma_f32_16x16x4_f32(false, af, false, b1,
                                                         (short)0, acc1, false, false);
            acc2 = __builtin_amdgcn_wmma_f32_16x16x4_f32(false, af, false, b2,
                                                         (short)0, acc2, false, false);
            acc3 = __builtin_amdgcn_wmma_f32_16x16x4_f32(false, af, false, b3,
                                                         (short)0, acc3, false, false);
        }
    };

    const int nk = K / BK;
    stage(0, 0);
    for (int ks = 0; ks < nk; ++ks) {
        wait_async0();          // this wave's DMA into current buffer done
        __syncthreads();        // every wave's DMA done; prior readers of next buf done
        if (ks + 1 < nk) stage((ks + 1) & 1, (ks + 1) * BK);  // overlap next panel DMA
        compute(ks & 1);
    }

    // C/D layout: VGPR r -> row (r + 8*laneHi), col = lane%16
    float* Cp = C + (size_t)(mbase + wave * 16 + laneHi * 8) * N + nbase + lanLo;
#pragma unroll
    for (int r = 0; r < 8; ++r) {
        Cp[(size_t)r * N + 0]  = acc0[r];
        Cp[(size_t)r * N + 16] = acc1[r];
        Cp[(size_t)r * N + 32] = acc2[r];
        Cp[(size_t)r * N + 48] = acc3[r];
    }
}

// ---------------------------------------------------------------------------
// S[j] = sum_i m[i,j]^2 via coalesced row-blocked partials + global f32 atomics
// (cols must be 1024). Followed by sqrt pass.
// ---------------------------------------------------------------------------
__global__ __launch_bounds__(256) void zero_S_kernel(float* __restrict__ S)
{
    S[blockIdx.x * 256 + threadIdx.x] = 0.f;
}

__global__ __launch_bounds__(256) void colnorm_partial_kernel(
    const float* __restrict__ m, float* __restrict__ S, int cols)
{
    const int r0 = blockIdx.x * 64;
    float s0 = 0.f, s1 = 0.f, s2 = 0.f, s3 = 0.f;
    for (int r = 0; r < 64; ++r) {
        const float* row = m + (size_t)(r0 + r) * cols;
        float v0 = row[threadIdx.x +   0];
        float v1 = row[threadIdx.x + 256];
        float v2 = row[threadIdx.x + 512];
        float v3 = row[threadIdx.x + 768];
        s0 += v0 * v0; s1 += v1 * v1; s2 += v2 * v2; s3 += v3 * v3;
    }
    atomicAdd(&S[threadIdx.x +   0], s0);   // global_atomic_add_f32
    atomicAdd(&S[threadIdx.x + 256], s1);
    atomicAdd(&S[threadIdx.x + 512], s2);
    atomicAdd(&S[threadIdx.x + 768], s3);
}

__global__ __launch_bounds__(256) void sqrt_S_kernel(float* __restrict__ S)
{
    int j = blockIdx.x * 256 + threadIdx.x;
    S[j] = sqrtf(S[j]) + 1e-20f;
}

// ---------------------------------------------------------------------------
// a = 0.5 * (U + L):  U ~ m with unit columns, L ~ unit-lower-trapezoid of x1
// ---------------------------------------------------------------------------
__global__ __launch_bounds__(256) void build_a_kernel(
    const float* __restrict__ m, const float* __restrict__ x1,
    const float* __restrict__ S, float* __restrict__ a, int rows, int cols)
{
    size_t idx = (size_t)blockIdx.x * blockDim.x + threadIdx.x;
    if (idx >= (size_t)rows * cols) return;
    int i = (int)(idx / cols);
    int j = (int)(idx % cols);
    float U  = m[idx] / S[j];
    float Lv = (j < i) ? x1[idx] : (j == i ? 1.0f : 0.0f);
    a[idx] = 0.5f * (U + Lv);
}

// ---------------------------------------------------------------------------
// b = max(Vh, Uu):  Vh ~ I, Uu ~ upper triangle of x1's leading CxC block
// ---------------------------------------------------------------------------
__global__ __launch_bounds__(256) void build_b_kernel(
    const float* __restrict__ x1, float* __restrict__ b, int cols)
{
    size_t idx = (size_t)blockIdx.x * blockDim.x + threadIdx.x;
    if (idx >= (size_t)cols * cols) return;
    int i = (int)(idx / cols);
    int j = (int)(idx % cols);
    float vh = (i == j) ? 1.0f : 0.0f;
    float uu = (j >= i) ? x1[idx] : 0.0f;
    b[idx] = fmaxf(vh, uu);
}

// ---------------------------------------------------------------------------
// out = t * S[j] + a   (P ~ I so P@a is the identity row-gather of a)
// ---------------------------------------------------------------------------
__global__ __launch_bounds__(256) void finalize_kernel(
    const float* __restrict__ t, const float* __restrict__ a,
    const float* __restrict__ S, float* __restrict__ out, int rows, int cols)
{
    size_t idx = (size_t)blockIdx.x * blockDim.x + threadIdx.x;
    if (idx >= (size_t)rows * cols) return;
    int j = (int)(idx % cols);
    out[idx] = t[idx] * S[j] + a[idx];
}

// ---------------------------------------------------------------------------
extern "C" void kernel_launch(void* const* d_in, const int* in_sizes, int n_in,
                              void* d_out, int out_size, void* d_ws, size_t ws_size,
                              hipStream_t stream)
{
    const int Brows = 4096;
    const int Ccols = 1024;

    const float* x0 = (const float*)d_in[0];
    const float* x1 = (const float*)d_in[1];
    const float* W  = (const float*)d_in[2];
    float* out = (float*)d_out;

    char* ws = (char*)d_ws;
    float* m = (float*)ws;  ws += (size_t)Brows * Ccols * sizeof(float);
    float* a = (float*)ws;  ws += (size_t)Brows * Ccols * sizeof(float);
    float* t = (float*)ws;  ws += (size_t)Brows * Ccols * sizeof(float);
    float* b = (float*)ws;  ws += (size_t)Ccols * Ccols * sizeof(float);
    float* S = (float*)ws;  ws += (size_t)Ccols * sizeof(float);
    (void)ws_size; (void)in_sizes; (void)n_in; (void)out_size;

    dim3 blk(256);
    dim3 ggrid(Ccols / BN, Brows / BM);   // N tiles x M tiles

    // m = x0 @ W
    gemm_wmma_f32<<<ggrid, blk, 0, stream>>>(x0, W, m, Brows, Ccols, Ccols);
    // S = column norms of m
    zero_S_kernel<<<Ccols / 256, blk, 0, stream>>>(S);
    colnorm_partial_kernel<<<Brows / 64, blk, 0, stream>>>(m, S, Ccols);
    sqrt_S_kernel<<<Ccols / 256, blk, 0, stream>>>(S);
    // a = 0.5*(U + L)
    build_a_kernel<<<(Brows * Ccols) / 256, blk, 0, stream>>>(m, x1, S, a, Brows, Ccols);
    // b = max(Vh, Uu)
    build_b_kernel<<<(Ccols * Ccols) / 256, blk, 0, stream>>>(x1, b, Ccols);
    // t = a @ b
    gemm_wmma_f32<<<ggrid, blk, 0, stream>>>(a, b, t, Brows, Ccols, Ccols);
    // out = t*S + a
    finalize_kernel<<<(Brows * Ccols) / 256, blk, 0, stream>>>(t, a, S, out, Brows, Ccols);
}